// GwaveGPU_31877247271369
// MI455X (gfx1250) — compile-verified
//
#include <hip/hip_runtime.h>
#include <math.h>

typedef __attribute__((ext_vector_type(2))) float v2f;
typedef __attribute__((ext_vector_type(8))) float v8f;

#define EPS_F       1e-10f
#define PI_F        3.14159265358979f
#define TAU_F       6.28318530717959f
#define INV_TAU_F   0.159154943091895f
#define LOG_CUTOFF  0.962423650119207f   // log(phi^2)

#define TJ    256   // j-tile staged in LDS
#define NTHR  256   // 8 waves * 32 lanes
#define IROWS 128   // 8 waves * 16 i-rows per block

__device__ __forceinline__ void precomp(float ell, float theta,
                                        float& lx, float& ly,
                                        float& sx, float& sy) {
    float c = __cosf(theta);
    float s = __sinf(theta);
    lx = ell + __logf(fabsf(c) + EPS_F);
    ly = ell + __logf(fabsf(s) + EPS_F);
    sx = (c > 0.f) ? 1.f : ((c < 0.f) ? -1.f : 0.f);
    sy = (s > 0.f) ? 1.f : ((s < 0.f) ? -1.f : 0.f);
}

// log|a-b| (same sign) or log(|a|+|b|) (opposite sign), in log space
__device__ __forceinline__ float log_axis(float la, float lb, bool same) {
    float hi = fmaxf(la, lb);
    float lo = fminf(la, lb);
    float d  = lo - hi;                      // <= 0
    float dc = fminf(d, -1e-7f);             // guard log(0) on diagonal
    float e  = __expf(same ? dc : d);
    float t  = same ? -e : e;
    return hi + __logf(1.0f + t);
}

__global__ void GwaveZero_kernel(float* __restrict__ out, int n) {
    int idx = blockIdx.x * blockDim.x + threadIdx.x;
    if (idx < n) out[idx] = 0.0f;
}

__global__ __launch_bounds__(NTHR)
void GwaveRepulse_kernel(const float* __restrict__ ellA,
                         const float* __restrict__ thetaA,
                         const float* __restrict__ sA,
                         const unsigned char* __restrict__ frozenA,
                         float* __restrict__ out,   // [2,N]: F_ell | F_theta
                         int N, int jchunk) {
    __shared__ float lxs[TJ], lys[TJ], sxs[TJ], sys[TJ];
    __shared__ float ths[TJ], els[TJ], sss[TJ];

    const int tid  = threadIdx.x;
    const int lane = tid & 31;
    const int wave = tid >> 5;
    const int ibase = blockIdx.x * IROWS + wave * 16;

    // per-lane i-row (lanes 0-15 and 16-31 hold the same 16 rows, per A layout)
    int  i      = ibase + (lane & 15);
    bool ivalid = (i < N);
    int  ic     = ivalid ? i : (N - 1);
    float ell_i = ellA[ic];
    float th_i  = thetaA[ic];
    float s_i   = (ivalid && (frozenA[ic] == 0)) ? sA[ic] : 0.0f; // ~frozen row mask
    float lx_i, ly_i, sx_i, sy_i;
    precomp(ell_i, th_i, lx_i, ly_i, sx_i, sy_i);

    const int jb   = blockIdx.y * jchunk;
    const int jend = min(jb + jchunk, N);

    v8f c_ell = {};              // WMMA accumulator: row-sums of force*d_ell
    v8f c_th  = {};              // WMMA accumulator: row-sums of force*d_theta
    v2f b_ones = {1.0f, 1.0f};   // all-ones B (4x16) -> D[m,n] = sum_k A[m,k] + C

    for (int j0 = jb; j0 < jend; j0 += TJ) {
        // ---- stage + precompute a j-tile into LDS (one j per thread) ----
        int   j  = j0 + tid;
        int   jc = (j < N) ? j : (N - 1);
        float ej = ellA[jc];
        float tj = thetaA[jc];
        float lxj, lyj, sxj, syj;
        precomp(ej, tj, lxj, lyj, sxj, syj);
        lxs[tid] = lxj;  lys[tid] = lyj;
        sxs[tid] = sxj;  sys[tid] = syj;
        ths[tid] = tj;   els[tid] = ej;
        sss[tid] = (j < jend) ? sA[jc] : 0.0f;   // zero pad -> zero force
        __syncthreads();

        // ---- inner loop: 4 j-columns per step, 2 pair-evals per lane ----
        for (int step = 0; step < TJ; step += 4) {
            v2f a_e, a_t;
#pragma unroll
            for (int kk = 0; kk < 2; ++kk) {
                // A-matrix 16x4 f32 layout: lanes 0-15 -> K=kk, lanes 16-31 -> K=2+kk
                int jj = step + ((lane >> 4) << 1) + kk;
                int jg = j0 + jj;

                float lx_j = lxs[jj], ly_j = lys[jj];
                float sx_j = sxs[jj], sy_j = sys[jj];
                float th_j = ths[jj], el_j = els[jj], s_j = sss[jj];

                float ldx = log_axis(lx_i, lx_j, (sx_i * sx_j) > 0.f);
                float ldy = log_axis(ly_i, ly_j, (sy_i * sy_j) > 0.f);

                // 0.5 * logaddexp(2*ldx, 2*ldy)
                float a2 = 2.0f * ldx, b2 = 2.0f * ldy;
                float h  = fmaxf(a2, b2), l = fminf(a2, b2);
                float ld = 0.5f * (h + __logf(1.0f + __expf(l - h)));

                bool  ok    = (ld <= LOG_CUTOFF) && (jg != i);
                float force = ok ? (s_i * s_j * __expf(-ld)) : 0.0f;

                float dth = th_j - th_i + PI_F;
                dth = dth - TAU_F * floorf(dth * INV_TAU_F);   // mod -> [0, tau)
                dth -= PI_F;

                a_e[kk] = force * (el_j - ell_i);
                a_t[kk] = force * dth;
            }
            // matrix-pipe j-reduction: C += A x ones  (exact f32 adds)
            c_ell = __builtin_amdgcn_wmma_f32_16x16x4_f32(
                        false, a_e, false, b_ones, (short)0, c_ell, false, false);
            c_th  = __builtin_amdgcn_wmma_f32_16x16x4_f32(
                        false, a_t, false, b_ones, (short)0, c_th, false, false);
        }
        __syncthreads();
    }

    // ---- writeout: D columns all hold the row-sum; use column 0 of each half ----
    // C/D layout: lanes 0-15, VGPR r -> M=r; lanes 16-31, VGPR r -> M=r+8
    if ((lane & 15) == 0) {
        int half = lane >> 4;  // 0 or 1
#pragma unroll
        for (int r = 0; r < 8; ++r) {
            int row = ibase + half * 8 + r;
            if (row < N) {
                __hip_atomic_fetch_add(&out[row],     c_ell[r],
                                       __ATOMIC_RELAXED, __HIP_MEMORY_SCOPE_AGENT);
                __hip_atomic_fetch_add(&out[N + row], c_th[r],
                                       __ATOMIC_RELAXED, __HIP_MEMORY_SCOPE_AGENT);
            }
        }
    }
}

extern "C" void kernel_launch(void* const* d_in, const int* in_sizes, int n_in,
                              void* d_out, int out_size, void* d_ws, size_t ws_size,
                              hipStream_t stream) {
    const float*         ell    = (const float*)d_in[0];
    const float*         theta  = (const float*)d_in[1];
    const float*         s      = (const float*)d_in[2];
    const unsigned char* frozen = (const unsigned char*)d_in[3];
    float*               out    = (float*)d_out;
    const int N = in_sizes[0];

    // zero the [2,N] output (atomically accumulated below)
    GwaveZero_kernel<<<(out_size + 255) / 256, 256, 0, stream>>>(out, out_size);

    const int JSPLIT = 8;                         // j-range split for occupancy
    const int jchunk = (N + JSPLIT - 1) / JSPLIT; // 512 for N=4096
    dim3 grid((N + IROWS - 1) / IROWS, JSPLIT);   // 32 x 8 = 256 blocks of 8 waves
    GwaveRepulse_kernel<<<grid, NTHR, 0, stream>>>(ell, theta, s, frozen,
                                                   out, N, jchunk);
}